// SS3D_76252849373721
// MI455X (gfx1250) — compile-verified
//
#include <hip/hip_runtime.h>
#include <hip/hip_bf16.h>

#define BATCH  2
#define DMDIM  96
#define DIDIM  192
#define NSTATE 16
#define RRANK  6
#define KDIR   6
#define LSEQ   4096
#define NPROJ  (RRANK + 2*NSTATE)   // 38
#define CHUNK  64                   // scan steps staged per async DMA round

typedef float v8f __attribute__((ext_vector_type(8)));
typedef float v2f __attribute__((ext_vector_type(2)));

// One V_WMMA_F32_16X16X4_F32 step: D = A(16x4) * B(4x16) + C
__device__ __forceinline__ v8f wmma_k4(v2f a, v2f b, v8f c) {
    return __builtin_amdgcn_wmma_f32_16x16x4_f32(
        /*neg_a=*/false, a, /*neg_b=*/false, b,
        /*c_mod=*/(short)0, c, /*reuse_a=*/false, /*reuse_b=*/false);
}

// gfx1250 async DMA: 16 bytes global -> LDS, tracked by ASYNCcnt.
// Generic pointers to LDS carry the wave-relative byte offset in addr[31:0].
__device__ __forceinline__ void async_copy16(const void* lds_dst, const float* gsrc) {
    uint32_t lds_off = (uint32_t)(uintptr_t)lds_dst;
    uint64_t gaddr   = (uint64_t)(uintptr_t)gsrc;
    asm volatile("global_load_async_to_lds_b128 %0, %1, off"
                 :: "v"(lds_off), "v"(gaddr) : "memory");
}

// Direction k maps sequence index l -> spatial offset (d,h,w) as a base-16
// digit permutation (k even) of l or of (L-1-l) (k odd, flipped sequences).
__device__ __forceinline__ int pos_map(int k, int l) {
    int lf = (k & 1) ? (LSEQ - 1 - l) : l;
    int i2 = lf >> 8, i1 = (lf >> 4) & 15, i0 = lf & 15;
    int kk = k >> 1;
    if (kk == 0) return lf;                         // seq (d,h,w)
    if (kk == 1) return (i1 << 8) | (i2 << 4) | i0; // l=(h,d,w) -> (d,h,w)
    return (i1 << 8) | (i0 << 4) | i2;              // l=(w,d,h) -> (d,h,w)
}

// ---------------------------------------------------------------------------
// 1) xz = x @ W_in^T ; split -> xe (conv layout [b,c,l]) and z ([b,l,c])
__global__ void k_inproj(const float* __restrict__ x, const float* __restrict__ Win,
                         float* __restrict__ xe, float* __restrict__ zb) {
    const int lane = threadIdx.x & 31, row = lane & 15, half = lane >> 4;
    const int m0 = blockIdx.x * 16, n0 = blockIdx.y * 16;
    const float* arow = x   + (size_t)(m0 + row) * DMDIM;
    const float* brow = Win + (size_t)(n0 + row) * DMDIM;
    v8f acc = {};
    #pragma unroll
    for (int kk = 0; kk < DMDIM; kk += 4) {
        int k = kk + half * 2;
        v2f a; a.x = arow[k]; a.y = arow[k + 1];
        v2f b; b.x = brow[k]; b.y = brow[k + 1];
        acc = wmma_k4(a, b, acc);
    }
    #pragma unroll
    for (int v = 0; v < 8; ++v) {
        int m = m0 + v + 8 * half, n = n0 + row;
        int b = m >> 12, l = m & (LSEQ - 1);
        float val = acc[v];
        if (n < DIDIM) xe[((size_t)b * DIDIM + n) * LSEQ + l] = val;
        else           zb[(size_t)m * DIDIM + (n - DIDIM)]    = val;
    }
}

// ---------------------------------------------------------------------------
// 2) depthwise 3x3x3 conv (pad 1) + bias + SiLU  -> xc [b,c,l]
__global__ void k_conv(const float* __restrict__ xe, const float* __restrict__ cw,
                       const float* __restrict__ cb, float* __restrict__ xc) {
    int idx = blockIdx.x * blockDim.x + threadIdx.x;   // < B*DI*L
    int l = idx & (LSEQ - 1);
    int c = (idx >> 12) % DIDIM;
    int b = idx / (DIDIM * LSEQ);
    int d = l >> 8, h = (l >> 4) & 15, w = l & 15;
    const float* src = xe + ((size_t)b * DIDIM + c) * LSEQ;
    const float* wk  = cw + c * 27;
    float sum = 0.f;
    #pragma unroll
    for (int dz = -1; dz <= 1; ++dz) {
        int zd = d + dz; if ((unsigned)zd >= 16u) continue;
        #pragma unroll
        for (int dy = -1; dy <= 1; ++dy) {
            int yh = h + dy; if ((unsigned)yh >= 16u) continue;
            #pragma unroll
            for (int dx = -1; dx <= 1; ++dx) {
                int xw = w + dx; if ((unsigned)xw >= 16u) continue;
                sum += src[(zd << 8) | (yh << 4) | xw] * wk[(dz+1)*9 + (dy+1)*3 + (dx+1)];
            }
        }
    }
    float v = sum + cb[c];
    xc[idx] = v / (1.f + __expf(-v));
}

// ---------------------------------------------------------------------------
// 3) x_dbl = per-direction projection DI->38, routed into dts / Bs / Cs.
__global__ void k_xdbl(const float* __restrict__ xc, const float* __restrict__ xw,
                       float* __restrict__ dts, float* __restrict__ Bs,
                       float* __restrict__ Cs) {
    const int lane = threadIdx.x & 31, row = lane & 15, half = lane >> 4;
    const int m0 = blockIdx.x * 16, n0 = blockIdx.y * 16, kdir = blockIdx.z;
    int m = m0 + row, b = m >> 12, l = m & (LSEQ - 1);
    int pos = pos_map(kdir, l);
    const float* arow = xc + (size_t)b * DIDIM * LSEQ + pos;   // + c*LSEQ
    int n = n0 + row;
    bool bok = (n < NPROJ);
    const float* brow = xw + ((size_t)kdir * NPROJ + (bok ? n : NPROJ - 1)) * DIDIM;
    v8f acc = {};
    #pragma unroll 4
    for (int cc = 0; cc < DIDIM; cc += 4) {
        int c = cc + half * 2;
        v2f a;  a.x = arow[(size_t)c * LSEQ]; a.y = arow[(size_t)(c + 1) * LSEQ];
        v2f bv; bv.x = bok ? brow[c] : 0.f;   bv.y = bok ? brow[c + 1] : 0.f;
        acc = wmma_k4(a, bv, acc);
    }
    #pragma unroll
    for (int v = 0; v < 8; ++v) {
        int mm = m0 + v + 8 * half, nn = n0 + row;
        int bb = mm >> 12, ll = mm & (LSEQ - 1);
        float val = acc[v];
        if (nn < RRANK)
            dts[(((size_t)bb * KDIR + kdir) * RRANK + nn) * LSEQ + ll] = val;
        else if (nn < RRANK + NSTATE)
            Bs[(((size_t)bb * KDIR + kdir) * NSTATE + (nn - RRANK)) * LSEQ + ll] = val;
        else if (nn < NPROJ)
            Cs[(((size_t)bb * KDIR + kdir) * NSTATE + (nn - RRANK - NSTATE)) * LSEQ + ll] = val;
    }
}

// ---------------------------------------------------------------------------
// 4) dt = softplus(dts @ dt_proj_w^T + b).  M=8192, N=192, K=6 (padded to 8).
__global__ void k_dtproj(const float* __restrict__ dts, const float* __restrict__ dw,
                         const float* __restrict__ db, float* __restrict__ dtb) {
    const int lane = threadIdx.x & 31, row = lane & 15, half = lane >> 4;
    const int m0 = blockIdx.x * 16, n0 = blockIdx.y * 16, kdir = blockIdx.z;
    int m = m0 + row, b = m >> 12, l = m & (LSEQ - 1);
    const float* arow = dts + (((size_t)b * KDIR + kdir) * RRANK) * LSEQ + l; // + r*LSEQ
    int n = n0 + row;
    const float* brow = dw + ((size_t)kdir * DIDIM + n) * RRANK;             // + r
    v8f acc = {};
    #pragma unroll
    for (int rr = 0; rr < 8; rr += 4) {
        int r = rr + half * 2;
        v2f a, bv;
        a.x  = (r     < RRANK) ? arow[(size_t)r * LSEQ]       : 0.f;
        a.y  = (r + 1 < RRANK) ? arow[(size_t)(r + 1) * LSEQ] : 0.f;
        bv.x = (r     < RRANK) ? brow[r]     : 0.f;
        bv.y = (r + 1 < RRANK) ? brow[r + 1] : 0.f;
        acc = wmma_k4(a, bv, acc);
    }
    #pragma unroll
    for (int v = 0; v < 8; ++v) {
        int mm = m0 + v + 8 * half, nn = n0 + row;
        int bb = mm >> 12, ll = mm & (LSEQ - 1);
        float val = acc[v] + db[kdir * DIDIM + nn];
        float sp = (val > 20.f) ? val : log1pf(__expf(val));
        dtb[(((size_t)bb * KDIR + kdir) * DIDIM + nn) * LSEQ + ll] = sp;
    }
}

// ---------------------------------------------------------------------------
// 5) selective scan with async-DMA double buffering of the dt/B/C streams.
//    Block = 256 threads = 16 channels x 16 states of one (b,k) slice.
//    64 chunks of 64 steps; chunk c+1 is DMA'd to LDS while chunk c runs.
__global__ void k_scan(const float* __restrict__ xc, const float* __restrict__ dtb,
                       const float* __restrict__ Bs, const float* __restrict__ Cs,
                       const float* __restrict__ A_logs, const float* __restrict__ Ds,
                       float* __restrict__ yacc) {
    __shared__ __align__(16) float dtS[2][16][68];   // padded stride 68: no bank conflicts
    __shared__ __align__(16) float BS [2][16][68];
    __shared__ __align__(16) float CS [2][16][68];

    const int tid = threadIdx.x;
    const int n = tid & 15, chl = tid >> 4;          // state index, local channel
    const int bk = blockIdx.x / 12, dg = blockIdx.x % 12;
    const int b = bk / KDIR, k = bk % KDIR, d = dg * 16 + chl;

    // staging roles: srow picks the channel/state row, scol the 16B segment
    const int srow = tid >> 4, scol = tid & 15;
    const float* dt_src = dtb + (((size_t)b * KDIR + k) * DIDIM  + dg * 16 + srow) * LSEQ;
    const float* B_src  = Bs  + (((size_t)b * KDIR + k) * NSTATE + srow) * LSEQ;
    const float* C_src  = Cs  + (((size_t)b * KDIR + k) * NSTATE + srow) * LSEQ;

    const float a  = -__expf(A_logs[((size_t)k * DIDIM + d) * NSTATE + n]);
    const float Dv = Ds[k * DIDIM + d];
    const float* up = xc + ((size_t)b * DIDIM + d) * LSEQ;
    float* yb = yacc + (size_t)b * LSEQ * DIDIM + d;

    auto stage = [&](int chunk) {
        const int bufb = chunk & 1;
        const int l0 = (chunk << 6) + (scol << 2);
        async_copy16(&dtS[bufb][srow][scol << 2], dt_src + l0);
        async_copy16(&BS [bufb][srow][scol << 2], B_src  + l0);
        async_copy16(&CS [bufb][srow][scol << 2], C_src  + l0);
    };

    stage(0);                                        // ASYNCcnt = 3 per wave
    float h = 0.f;
    for (int c = 0; c < LSEQ / CHUNK; ++c) {
        if (c + 1 < LSEQ / CHUNK) {
            stage(c + 1);                            // keep next chunk in flight
            asm volatile("s_wait_asynccnt 0x3" ::: "memory");  // chunk c landed
        } else {
            asm volatile("s_wait_asynccnt 0x0" ::: "memory");
        }
        __syncthreads();

        const int bufb = c & 1;
        const float* dtRow = &dtS[bufb][chl][0];
        const float* BRow  = &BS [bufb][n][0];
        const float* CRow  = &CS [bufb][n][0];
        #pragma unroll 4
        for (int j = 0; j < CHUNK; ++j) {
            int l = (c << 6) + j;
            int pos = pos_map(k, l);
            float dtv = dtRow[j], Bt = BRow[j], Ct = CRow[j], u = up[pos];
            h = __expf(dtv * a) * h + (dtv * u) * Bt;
            float yv = h * Ct;
            #pragma unroll
            for (int off = 8; off; off >>= 1) yv += __shfl_xor(yv, off, 16);
            if (n == 0) atomicAdd(yb + (size_t)pos * DIDIM, yv + Dv * u);
        }
        __syncthreads();                             // all reads done before buffer reuse
    }
}

// ---------------------------------------------------------------------------
// 6) LayerNorm over DI + SiLU(z) gate. One wave per (b,l) row; 6 elems/lane.
__global__ void k_ln_gate(const float* __restrict__ y, const float* __restrict__ zb,
                          const float* __restrict__ lw, const float* __restrict__ lb,
                          float* __restrict__ yg) {
    const int wave = threadIdx.x >> 5, lane = threadIdx.x & 31;
    const int rowid = blockIdx.x * (blockDim.x >> 5) + wave;  // < 8192
    const float* yr = y  + (size_t)rowid * DIDIM;
    const float* zr = zb + (size_t)rowid * DIDIM;
    float vals[6], s = 0.f, s2 = 0.f;
    #pragma unroll
    for (int i = 0; i < 6; ++i) {
        float t = yr[lane * 6 + i]; vals[i] = t; s += t; s2 += t * t;
    }
    #pragma unroll
    for (int off = 16; off; off >>= 1) {
        s  += __shfl_xor(s,  off, 32);
        s2 += __shfl_xor(s2, off, 32);
    }
    float mu = s * (1.f / DIDIM);
    float var = s2 * (1.f / DIDIM) - mu * mu;
    float rstd = rsqrtf(var + 1e-5f);
    #pragma unroll
    for (int i = 0; i < 6; ++i) {
        int c = lane * 6 + i;
        float zn = zr[c];
        float gate = zn / (1.f + __expf(-zn));
        yg[(size_t)rowid * DIDIM + c] = ((vals[i] - mu) * rstd * lw[c] + lb[c]) * gate;
    }
}

// ---------------------------------------------------------------------------
// 7) out = yg @ W_out^T.  M=8192, N=96, K=192.
__global__ void k_outproj(const float* __restrict__ yg, const float* __restrict__ Wout,
                          float* __restrict__ out) {
    const int lane = threadIdx.x & 31, row = lane & 15, half = lane >> 4;
    const int m0 = blockIdx.x * 16, n0 = blockIdx.y * 16;
    const float* arow = yg   + (size_t)(m0 + row) * DIDIM;
    const float* brow = Wout + (size_t)(n0 + row) * DIDIM;
    v8f acc = {};
    #pragma unroll 4
    for (int cc = 0; cc < DIDIM; cc += 4) {
        int c = cc + half * 2;
        v2f a; a.x = arow[c]; a.y = arow[c + 1];
        v2f b; b.x = brow[c]; b.y = brow[c + 1];
        acc = wmma_k4(a, b, acc);
    }
    #pragma unroll
    for (int v = 0; v < 8; ++v) {
        int mm = m0 + v + 8 * half, nn = n0 + row;
        out[(size_t)mm * DMDIM + nn] = acc[v];
    }
}

// ---------------------------------------------------------------------------
extern "C" void kernel_launch(void* const* d_in, const int* in_sizes, int n_in,
                              void* d_out, int out_size, void* d_ws, size_t ws_size,
                              hipStream_t stream) {
    const float* x        = (const float*)d_in[0];
    const float* W_in     = (const float*)d_in[1];
    const float* conv_w   = (const float*)d_in[2];
    const float* conv_b   = (const float*)d_in[3];
    const float* x_proj_w = (const float*)d_in[4];
    const float* dt_w     = (const float*)d_in[5];
    const float* dt_b     = (const float*)d_in[6];
    const float* A_logs   = (const float*)d_in[7];
    const float* Ds       = (const float*)d_in[8];
    const float* ln_w     = (const float*)d_in[9];
    const float* ln_b     = (const float*)d_in[10];
    const float* W_out    = (const float*)d_in[11];
    float* out = (float*)d_out;

    float* ws = (float*)d_ws;
    size_t off = 0;
    float* xe   = ws + off; off += (size_t)BATCH * DIDIM * LSEQ;
    float* zb   = ws + off; off += (size_t)BATCH * LSEQ * DIDIM;
    float* xc   = ws + off; off += (size_t)BATCH * DIDIM * LSEQ;
    float* dts  = ws + off; off += (size_t)BATCH * KDIR * RRANK * LSEQ;
    float* Bsb  = ws + off; off += (size_t)BATCH * KDIR * NSTATE * LSEQ;
    float* Csb  = ws + off; off += (size_t)BATCH * KDIR * NSTATE * LSEQ;
    float* dtb  = ws + off; off += (size_t)BATCH * KDIR * DIDIM * LSEQ;
    float* yacc = ws + off; off += (size_t)BATCH * LSEQ * DIDIM;
    float* yg   = ws + off; off += (size_t)BATCH * LSEQ * DIDIM;

    const int M = BATCH * LSEQ;  // 8192

    hipMemsetAsync(yacc, 0, (size_t)BATCH * LSEQ * DIDIM * sizeof(float), stream);

    k_inproj <<<dim3(M / 16, (2 * DIDIM) / 16), 32, 0, stream>>>(x, W_in, xe, zb);
    k_conv   <<<dim3((BATCH * DIDIM * LSEQ) / 256), 256, 0, stream>>>(xe, conv_w, conv_b, xc);
    k_xdbl   <<<dim3(M / 16, 3, KDIR), 32, 0, stream>>>(xc, x_proj_w, dts, Bsb, Csb);
    k_dtproj <<<dim3(M / 16, DIDIM / 16, KDIR), 32, 0, stream>>>(dts, dt_w, dt_b, dtb);
    k_scan   <<<dim3((BATCH * KDIR * DIDIM) / 16), 256, 0, stream>>>(xc, dtb, Bsb, Csb,
                                                                     A_logs, Ds, yacc);
    k_ln_gate<<<dim3(M / 8), 256, 0, stream>>>(yacc, zb, ln_w, ln_b, yg);
    k_outproj<<<dim3(M / 16, DMDIM / 16), 32, 0, stream>>>(yg, W_out, out);
}